// LGCN_80041010528459
// MI455X (gfx1250) — compile-verified
//
#include <hip/hip_runtime.h>
#include <math.h>

#define F 128
#define EPSF 1e-5f
#define USE_ASYNC_LDS 1

typedef __attribute__((ext_vector_type(2))) float v2f;
typedef __attribute__((ext_vector_type(8))) float v8f;

__device__ __forceinline__ void fatomic_add(float* p, float v) {
  unsafeAtomicAdd(p, v);   // lowers to global_atomic_add_f32 on gfx1250
}

// ---------------------------------------------------------------- utilities
__global__ __launch_bounds__(256) void k_fill(float* p, float v, size_t n) {
  size_t i = (size_t)blockIdx.x * blockDim.x + threadIdx.x;
  if (i < n) p[i] = v;
}

__global__ __launch_bounds__(256) void k_degree(const int* __restrict__ src,
                                                const int* __restrict__ dst,
                                                int E, float* dout, float* din) {
  int e = blockIdx.x * blockDim.x + threadIdx.x;
  if (e >= E) return;
  fatomic_add(&dout[src[e]], 1.0f);
  fatomic_add(&din[dst[e]], 1.0f);
}

__global__ __launch_bounds__(256) void k_rsqrt(float* p, int n) {
  int i = blockIdx.x * blockDim.x + threadIdx.x;
  if (i < n) p[i] = rsqrtf(p[i]);
}

// Y[row,:] = X[row,:] * s[row]   (float4 per thread; 32 threads per row)
__global__ __launch_bounds__(256) void k_scale_rows(const float* __restrict__ X,
                                                    const float* __restrict__ s,
                                                    float* __restrict__ Y, int N) {
  size_t gid = (size_t)blockIdx.x * blockDim.x + threadIdx.x;
  if (gid >= (size_t)N * 32) return;
  int row = (int)(gid >> 5);
  float sc = s[row];
  float4 v = *(const float4*)(X + gid * 4);
  v.x *= sc; v.y *= sc; v.z *= sc; v.w *= sc;
  *(float4*)(Y + gid * 4) = v;
}

// Y[e,:] = X[idx[e],:] * s[e]   (wave per row)
__global__ __launch_bounds__(256) void k_gather_scale(const float* __restrict__ X,
                                                      const int* __restrict__ idx,
                                                      const float* __restrict__ s,
                                                      float* __restrict__ Y, int E) {
  int wave = threadIdx.x >> 5, lane = threadIdx.x & 31;
  int e = blockIdx.x * 8 + wave;
  if (e >= E) return;
  int g = idx[e];
  float sc = s[e];
  float4 v = *(const float4*)(X + (size_t)g * F + lane * 4);
  v.x *= sc; v.y *= sc; v.z *= sc; v.w *= sc;
  *(float4*)(Y + (size_t)e * F + lane * 4) = v;
}

// Y[sidx[e],:] += X[g,:] where g = gidx ? gidx[e] : e   (wave per edge)
__global__ __launch_bounds__(256) void k_scatter_add(const float* __restrict__ X,
                                                     const int* __restrict__ gidx,
                                                     const int* __restrict__ sidx,
                                                     float* __restrict__ Y, int E) {
  int wave = threadIdx.x >> 5, lane = threadIdx.x & 31;
  int e = blockIdx.x * 8 + wave;
  if (e >= E) return;
  int g = gidx ? gidx[e] : e;
  int d = sidx[e];
  float4 v = *(const float4*)(X + (size_t)g * F + lane * 4);
  float* yp = Y + (size_t)d * F + lane * 4;
  fatomic_add(yp + 0, v.x);
  fatomic_add(yp + 1, v.y);
  fatomic_add(yp + 2, v.z);
  fatomic_add(yp + 3, v.w);
}

// ---------------------------------------------------------------- WMMA GEMM
// out[n,f] (+)= sc[f] * ( sum_k A[n,k]*W[k,f] + bias[f] )
// W (64KB) is staged once per block into LDS via the CDNA5 async global->LDS
// path (ASYNCcnt + barrier), then each wave computes one 16-row tile:
// 8 column blocks x 32 v_wmma_f32_16x16x4_f32, B fragments fed from LDS.
// Two independent accumulator chains hide the WMMA->WMMA co-exec hazard.
__global__ __launch_bounds__(256) void k_gemm128(const float* __restrict__ A,
                                                 const float* __restrict__ W,
                                                 const float* __restrict__ bias,
                                                 const float* __restrict__ sc,
                                                 float* __restrict__ out,
                                                 int ntiles, int accumulate) {
  __shared__ float ldsW[F * F];

  // LDS byte offset of ldsW: flat shared address low 32 bits == LDS offset
  // (ISA flat-aperture mapping); the ptrtoint also escapes ldsW so the asm's
  // memory clobber is seen as defining it.
  unsigned ldsbase = (unsigned)(uintptr_t)(void*)ldsW;

  // ---- stage W: 256 threads x 16 x 16B = 64KB
#if USE_ASYNC_LDS
  for (int i = threadIdx.x; i < (F * F) / 4; i += 256) {
    unsigned loff = ldsbase + (unsigned)(i * 16);
    const float* gp = W + (size_t)i * 4;
    asm volatile("global_load_async_to_lds_b128 %0, %1, off"
                 :: "v"(loff), "v"(gp) : "memory");
  }
  asm volatile("s_wait_asynccnt 0x0" ::: "memory");
#else
  for (int i = threadIdx.x; i < (F * F) / 4; i += 256) {
    *(float4*)&ldsW[i * 4] = *(const float4*)(W + (size_t)i * 4);
  }
#endif
  __syncthreads();

  int wave = threadIdx.x >> 5, lane = threadIdx.x & 31;
  int tile = blockIdx.x * 8 + wave;
  if (tile < ntiles) {                   // wave-uniform: EXEC stays all-ones
    int row0 = tile * 16;
    int m = lane & 15;
    int khalf = (lane >> 4) << 1;        // 0 for lanes 0-15, 2 for lanes 16-31
    const float* arow = A + (size_t)(row0 + m) * F + khalf;
    for (int nb = 0; nb < 8; ++nb) {
      int col = (nb << 4) + m;
      const float* wl = ldsW + (size_t)khalf * F + col;
      v8f acc0 = {0.f, 0.f, 0.f, 0.f, 0.f, 0.f, 0.f, 0.f};
      v8f acc1 = {0.f, 0.f, 0.f, 0.f, 0.f, 0.f, 0.f, 0.f};
#pragma unroll
      for (int k0 = 0; k0 < F; k0 += 8) {
        v2f a0 = *(const v2f*)(arow + k0);             // A[m, k0+khalf .. +1]
        v2f b0;
        b0.x = wl[(size_t)k0 * F];                     // W[k0+khalf,   col]
        b0.y = wl[(size_t)(k0 + 1) * F];               // W[k0+khalf+1, col]
        acc0 = __builtin_amdgcn_wmma_f32_16x16x4_f32(false, a0, false, b0,
                                                     (short)0, acc0, false, false);
        v2f a1 = *(const v2f*)(arow + k0 + 4);
        v2f b1;
        b1.x = wl[(size_t)(k0 + 4) * F];
        b1.y = wl[(size_t)(k0 + 5) * F];
        acc1 = __builtin_amdgcn_wmma_f32_16x16x4_f32(false, a1, false, b1,
                                                     (short)0, acc1, false, false);
      }
      v8f acc = acc0 + acc1;
      float bb = bias[col], ssc = sc[col];
      int rbase = row0 + ((lane & 16) ? 8 : 0);
#pragma unroll
      for (int v = 0; v < 8; ++v) {
        size_t idx = (size_t)(rbase + v) * F + col;
        float val = (acc[v] + bb) * ssc;
        if (accumulate) val += out[idx];
        out[idx] = val;
      }
    }
  }
}

// ---------------------------------------------------------------- norm chain
// In-place LayerNorm over F=128 + ReLU on upper half. One wave per row.
__global__ __launch_bounds__(256) void k_layernorm(float* __restrict__ R,
                                                   const float* __restrict__ g,
                                                   const float* __restrict__ b,
                                                   int N) {
  int wave = threadIdx.x >> 5, lane = threadIdx.x & 31;
  int row = blockIdx.x * 8 + wave;
  if (row >= N) return;
  float* p = R + (size_t)row * F + lane * 4;
  float4 v = *(const float4*)p;
  float s = v.x + v.y + v.z + v.w;
#pragma unroll
  for (int msk = 16; msk >= 1; msk >>= 1) s += __shfl_xor(s, msk, 32);
  float mu = s * (1.0f / F);
  float dx = v.x - mu, dy = v.y - mu, dz = v.z - mu, dw = v.w - mu;
  float q = dx * dx + dy * dy + dz * dz + dw * dw;
#pragma unroll
  for (int msk = 16; msk >= 1; msk >>= 1) q += __shfl_xor(q, msk, 32);
  float inv = rsqrtf(q * (1.0f / F) + EPSF);
  int c = lane * 4;
  float4 o;
  o.x = dx * inv * g[c + 0] + b[c + 0];
  o.y = dy * inv * g[c + 1] + b[c + 1];
  o.z = dz * inv * g[c + 2] + b[c + 2];
  o.w = dw * inv * g[c + 3] + b[c + 3];
  if (lane >= 16) {                      // columns >= 64: ReLU half
    o.x = fmaxf(o.x, 0.f); o.y = fmaxf(o.y, 0.f);
    o.z = fmaxf(o.z, 0.f); o.w = fmaxf(o.w, 0.f);
  }
  *(float4*)p = o;
}

// Per-feature sum / sum-of-squares over all N rows (grid-strided, reg accum).
__global__ __launch_bounds__(128) void k_colstats(const float* __restrict__ R,
                                                  float* __restrict__ sum,
                                                  float* __restrict__ sq, int N) {
  int c = threadIdx.x;
  float s = 0.f, ss = 0.f;
  for (int r = blockIdx.x; r < N; r += gridDim.x) {
    __builtin_prefetch(R + (size_t)(r + gridDim.x) * F + c, 0, 1);
    float v = R[(size_t)r * F + c];
    s += v;
    ss += v * v;
  }
  fatomic_add(&sum[c], s);
  fatomic_add(&sq[c], ss);
}

// BatchNorm epilogue -> d_out
__global__ __launch_bounds__(256) void k_bn(const float* __restrict__ R,
                                            const float* __restrict__ sum,
                                            const float* __restrict__ sq,
                                            const float* __restrict__ g,
                                            const float* __restrict__ b,
                                            float* __restrict__ out, int N) {
  size_t gid = (size_t)blockIdx.x * blockDim.x + threadIdx.x;
  if (gid >= (size_t)N * 32) return;
  int c0 = (int)(gid & 31) * 4;
  float invN = 1.0f / (float)N;
  float4 v = *(const float4*)(R + gid * 4);
  float4 o;
  {
    float mu = sum[c0 + 0] * invN, var = sq[c0 + 0] * invN - mu * mu;
    o.x = (v.x - mu) * rsqrtf(var + EPSF) * g[c0 + 0] + b[c0 + 0];
  }
  {
    float mu = sum[c0 + 1] * invN, var = sq[c0 + 1] * invN - mu * mu;
    o.y = (v.y - mu) * rsqrtf(var + EPSF) * g[c0 + 1] + b[c0 + 1];
  }
  {
    float mu = sum[c0 + 2] * invN, var = sq[c0 + 2] * invN - mu * mu;
    o.z = (v.z - mu) * rsqrtf(var + EPSF) * g[c0 + 2] + b[c0 + 2];
  }
  {
    float mu = sum[c0 + 3] * invN, var = sq[c0 + 3] * invN - mu * mu;
    o.w = (v.w - mu) * rsqrtf(var + EPSF) * g[c0 + 3] + b[c0 + 3];
  }
  *(float4*)(out + gid * 4) = o;
}

// ---------------------------------------------------------------- host side
static inline unsigned cdiv(size_t a, size_t b) { return (unsigned)((a + b - 1) / b); }

extern "C" void kernel_launch(void* const* d_in, const int* in_sizes, int n_in,
                              void* d_out, int out_size, void* d_ws, size_t ws_size,
                              hipStream_t stream) {
  const float* h0 = (const float*)d_in[0];
  const float* h1 = (const float*)d_in[1];
  const float* h2 = (const float*)d_in[2];
  const int* src0 = (const int*)d_in[3];
  const int* dst0 = (const int*)d_in[4];
  const int* src1 = (const int*)d_in[5];
  const int* dst1 = (const int*)d_in[6];
  const int* src2 = (const int*)d_in[7];
  const int* dst2 = (const int*)d_in[8];

  const int N0 = in_sizes[0] / F, N1 = in_sizes[1] / F, N2 = in_sizes[2] / F;
  const int E0 = in_sizes[3], E1 = in_sizes[5], E2 = in_sizes[7];

  // params: 3 cores (top_down, both, bottom_up) x 11 tensors in dict order
  enum { iWc = 0, ibc, iWf, ibf, iconv_w, itd_w, ibu_w, iln_g, iln_b, ibn_g, ibn_b };
  auto P = [&](int core, int j) { return (const float*)d_in[9 + core * 11 + j]; };

  // workspace layout
  size_t cap = (size_t)N2 * F;
  float* A      = (float*)d_ws;       // conv input (hs) buffer
  float* AG     = A + cap;            // aggregation buffer
  float* R      = AG + cap;           // accumulated r per level
  float* degout = R + cap;            // N floats
  float* degin  = degout + N2;        // N floats
  float* csum   = degin + N2;         // F floats
  float* csq    = csum + F;           // F floats

  float* out  = (float*)d_out;
  float* out0 = out;
  float* out1 = out0 + (size_t)N0 * F;
  float* out2 = out1 + (size_t)N1 * F;

  auto fill = [&](float* p, float v, size_t n) {
    k_fill<<<cdiv(n, 256), 256, 0, stream>>>(p, v, n);
  };
  auto degrees = [&](int N, int E, const int* src, const int* dst) {
    fill(degout, 1.0f, (size_t)N);
    fill(degin, 1.0f, (size_t)N);
    k_degree<<<cdiv(E, 256), 256, 0, stream>>>(src, dst, E, degout, degin);
    k_rsqrt<<<cdiv(N, 256), 256, 0, stream>>>(degout, N);
    k_rsqrt<<<cdiv(N, 256), 256, 0, stream>>>(degin, N);
  };
  // A already holds hs; runs: agg = hs + scatter(hs[src]->dst); agg *= degin;
  // R (+)= sc * (agg @ W + b)
  auto conv = [&](int N, int E, const int* src, const int* dst,
                  const float* Wm, const float* bv, const float* scv, int accF) {
    hipMemcpyAsync(AG, A, (size_t)N * F * sizeof(float),
                   hipMemcpyDeviceToDevice, stream);                  // self loop
    k_scatter_add<<<cdiv(E, 8), 256, 0, stream>>>(A, src, dst, AG, E);
    k_scale_rows<<<cdiv((size_t)N * 32, 256), 256, 0, stream>>>(AG, degin, AG, N);
    k_gemm128<<<cdiv((size_t)(N / 16), 8), 256, 0, stream>>>(AG, Wm, bv, scv, R,
                                                             N / 16, accF);
  };
  auto normchain = [&](int N, int core, float* op) {
    k_layernorm<<<cdiv(N, 8), 256, 0, stream>>>(R, P(core, iln_g), P(core, iln_b), N);
    hipMemsetAsync(csum, 0, F * sizeof(float), stream);
    hipMemsetAsync(csq, 0, F * sizeof(float), stream);
    k_colstats<<<512, 128, 0, stream>>>(R, csum, csq, N);
    k_bn<<<cdiv((size_t)N * 32, 256), 256, 0, stream>>>(R, csum, csq,
                                                        P(core, ibn_g),
                                                        P(core, ibn_b), op, N);
  };

  // ---------------- level 0 (top_down core) ----------------
  degrees(N0, E0, src0, dst0);
  // conv(h0) * conv_w
  k_scale_rows<<<cdiv((size_t)N0 * 32, 256), 256, 0, stream>>>(h0, degout, A, N0);
  conv(N0, E0, src0, dst0, P(0, iWc), P(0, ibc), P(0, iconv_w), 0);
  // fused0 = segsum(h1, dst0);  conv(fused0) * td_w
  hipMemsetAsync(A, 0, (size_t)N0 * F * sizeof(float), stream);
  k_scatter_add<<<cdiv(E0, 8), 256, 0, stream>>>(h1, nullptr, dst0, A, E0);
  k_scale_rows<<<cdiv((size_t)N0 * 32, 256), 256, 0, stream>>>(A, degout, A, N0);
  conv(N0, E0, src0, dst0, P(0, iWf), P(0, ibf), P(0, itd_w), 1);
  normchain(N0, 0, out0);

  // ---------------- level 1 (both core) ----------------
  degrees(N1, E1, src1, dst1);
  // conv(h1) * conv_w
  k_scale_rows<<<cdiv((size_t)N1 * 32, 256), 256, 0, stream>>>(h1, degout, A, N1);
  conv(N1, E1, src1, dst1, P(1, iWc), P(1, ibc), P(1, iconv_w), 0);
  // fused_bu = h0[dst0];  conv(fused_bu) * bu_w  (gather fused with degout scale)
  k_gather_scale<<<cdiv(N1, 8), 256, 0, stream>>>(h0, dst0, degout, A, N1);
  conv(N1, E1, src1, dst1, P(1, iWf), P(1, ibf), P(1, ibu_w), 1);
  // fused_td = segsum(h2, dst1);  conv(fused_td) * td_w
  hipMemsetAsync(A, 0, (size_t)N1 * F * sizeof(float), stream);
  k_scatter_add<<<cdiv(E1, 8), 256, 0, stream>>>(h2, nullptr, dst1, A, E1);
  k_scale_rows<<<cdiv((size_t)N1 * 32, 256), 256, 0, stream>>>(A, degout, A, N1);
  conv(N1, E1, src1, dst1, P(1, iWf), P(1, ibf), P(1, itd_w), 1);
  normchain(N1, 1, out1);

  // ---------------- level 2 (bottom_up core) ----------------
  degrees(N2, E2, src2, dst2);
  // conv(h2) * conv_w
  k_scale_rows<<<cdiv((size_t)N2 * 32, 256), 256, 0, stream>>>(h2, degout, A, N2);
  conv(N2, E2, src2, dst2, P(2, iWc), P(2, ibc), P(2, iconv_w), 0);
  // fused2 = h1[dst1];  conv(fused2) * bu_w
  k_gather_scale<<<cdiv(N2, 8), 256, 0, stream>>>(h1, dst1, degout, A, N2);
  conv(N2, E2, src2, dst2, P(2, iWf), P(2, ibf), P(2, ibu_w), 1);
  normchain(N2, 2, out2);
}